// StatefulRecurrent_3650722201841
// MI455X (gfx1250) — compile-verified
//
#include <hip/hip_runtime.h>
#include <stdint.h>

// Problem constants (match reference)
#define B_SZ 16
#define T_SZ 4096
#define D_SZ 512

// Tiling
#define DBLK   8                    // d-channels per block
#define CHUNK  32                   // timesteps per chunk
#define NCHUNK (T_SZ / CHUNK)       // 128 chunks
#define LANES  4                    // lane-groups per chunk (2 channels each)
#define THREADS (NCHUNK * LANES)    // 512 threads
#define ROWW   9                    // LDS row stride in dwords (8 data + 1 TDM pad)

typedef unsigned int u32x4 __attribute__((ext_vector_type(4)));
typedef int          i32x8 __attribute__((ext_vector_type(8)));
typedef int          i32x4 __attribute__((ext_vector_type(4)));
typedef float        f32x4 __attribute__((ext_vector_type(4)));
typedef float        f32x2 __attribute__((ext_vector_type(2)));

__global__ __launch_bounds__(THREADS, 1)
void StatefulRecurrent_scan_kernel(const float* __restrict__ x,
                                   const float* __restrict__ Ar,
                                   const float* __restrict__ Ai,
                                   float* __restrict__ out) {
    // LDS: x tile (4096 rows x 9 dwords = 144KB) + chunk states (8KB)
    __shared__ float xbuf[T_SZ * ROWW];
    __shared__ float fst[NCHUNK * DBLK * 2];

    const int b    = blockIdx.x >> 6;       // 0..15
    const int dblk = blockIdx.x & 63;       // 0..63
    const int d0   = dblk * DBLK;

    const int tid = threadIdx.x;
    const int l   = tid & (LANES - 1);      // 0..3
    const int c   = tid >> 2;               // 0..127 (chunk id)
    const int ch0 = 2 * l;                  // first of 2 channels owned

    const float* xb = x + ((size_t)b * T_SZ) * D_SZ + d0;   // x[b][0][d0]

    // ---------------- TDM: DMA the block's x tile into LDS (wave 0) ----------
    if (tid < 32) {
        const uint32_t ldsoff = (uint32_t)(uintptr_t)(&xbuf[0]); // low32 of LDS ptr = wave-relative offset
        const uint64_t ga     = (uint64_t)(uintptr_t)xb;

        u32x4 g0;
        g0[0] = 1u;                                             // count=1, user mode
        g0[1] = ldsoff;                                         // lds_addr
        g0[2] = (uint32_t)ga;                                   // global_addr[31:0]
        g0[3] = (uint32_t)((ga >> 32) & 0x01FFFFFFu)            // global_addr[56:32]
              | (2u << 30);                                     // type=2 (image)

        i32x8 g1;
        g1[0] = (int)((2u << 16)        // data_size = 4B
                    | (1u << 20)        // pad_enable
                    | (2u << 22));      // pad_interval: code 2 = every 8 dwords
                                        // pad_amount:  code 0 = 1 dword
        g1[1] = (int)(512u  << 16);     // tensor_dim0 = 512 (lo16)
        g1[2] = (int)(4096u << 16);     // tensor_dim0 hi=0 | tensor_dim1 lo = 4096
        g1[3] = (int)(8u    << 16);     // tensor_dim1 hi=0 | tile_dim0 = 8
        g1[4] = (int)(4096u);           // tile_dim1 = 4096 | tile_dim2 = 0
        g1[5] = (int)(512u);            // tensor_dim0_stride = 512 (lo32)
        g1[6] = 0;                      // stride hi / dim1_stride lo
        g1[7] = 0;                      // dim1_stride hi

        i32x4 gz4 = {0, 0, 0, 0};               // groups 2/3 unused (2D tensor)
        i32x8 gz8 = {0, 0, 0, 0, 0, 0, 0, 0};   // extra group (clang-23 6-arg form)
        __builtin_amdgcn_tensor_load_to_lds(g0, g1, gz4, gz4, gz8, 0);
    }

    // ---------------- Pass 1: per-chunk scan from zero (x read from global) ---
    const float ar0 = Ar[d0 + ch0],     ai0 = Ai[d0 + ch0];
    const float ar1 = Ar[d0 + ch0 + 1], ai1 = Ai[d0 + ch0 + 1];
    {
        const float* p = xb + (size_t)(c * CHUNK) * D_SZ + ch0;
        float sr0 = 0.f, si0 = 0.f, sr1 = 0.f, si1 = 0.f;
        #pragma unroll 4
        for (int j = 0; j < CHUNK; ++j) {
            const float x0 = p[0];
            const float x1 = p[1];
            const float nr0 = fmaf(sr0, ar0, fmaf(-si0, ai0, x0));
            const float ni0 = fmaf(sr0, ai0, si0 * ar0);
            const float nr1 = fmaf(sr1, ar1, fmaf(-si1, ai1, x1));
            const float ni1 = fmaf(sr1, ai1, si1 * ar1);
            sr0 = nr0; si0 = ni0; sr1 = nr1; si1 = ni1;
            p += D_SZ;
        }
        f32x4* f4 = (f32x4*)&fst[(c * DBLK + ch0) * 2];
        *f4 = f32x4{sr0, si0, sr1, si1};
    }
    __syncthreads();

    // ---------------- Pass 2: in-place chunk-state scan (1 thread / channel) --
    if (tid < DBLK) {
        const int ch = tid;
        float mr = Ar[d0 + ch], mi = Ai[d0 + ch];
        #pragma unroll
        for (int q = 0; q < 5; ++q) {          // M = A^32
            const float r2 = mr * mr - mi * mi;
            const float i2 = 2.f * mr * mi;
            mr = r2; mi = i2;
        }
        float cr = 0.f, ci = 0.f;              // incoming state for chunk 0
        for (int cc = 0; cc < NCHUNK; ++cc) {
            f32x2* f2 = (f32x2*)&fst[(cc * DBLK + ch) * 2];
            const f32x2 f = *f2;               // local final of chunk cc
            *f2 = f32x2{cr, ci};               // replace with incoming state
            const float nr = fmaf(mr, cr, fmaf(-mi, ci, f[0]));
            const float ni = fmaf(mi, cr, fmaf( mr, ci, f[1]));
            cr = nr; ci = ni;
        }
    }
    // TDM must be complete before pass 3 reads xbuf (issuing wave waits, then barrier)
    if (tid < 32) {
        __builtin_amdgcn_s_wait_tensorcnt(0);
    }
    __syncthreads();

    // ---------------- Pass 3: corrected re-scan from LDS x, stream out --------
    {
        const f32x4 s = *(const f32x4*)&fst[(c * DBLK + ch0) * 2];
        float sr0 = s[0], si0 = s[1], sr1 = s[2], si1 = s[3];
        float* op = out + (((size_t)b * T_SZ + (size_t)c * CHUNK) * D_SZ + d0 + ch0) * 2;
        const int row0 = c * CHUNK;
        #pragma unroll 4
        for (int j = 0; j < CHUNK; ++j) {
            const int base = (row0 + j) * ROWW + ch0;
            const float x0 = xbuf[base];
            const float x1 = xbuf[base + 1];
            const float nr0 = fmaf(sr0, ar0, fmaf(-si0, ai0, x0));
            const float ni0 = fmaf(sr0, ai0, si0 * ar0);
            const float nr1 = fmaf(sr1, ar1, fmaf(-si1, ai1, x1));
            const float ni1 = fmaf(sr1, ai1, si1 * ar1);
            sr0 = nr0; si0 = ni0; sr1 = nr1; si1 = ni1;
            const f32x4 o = {nr0, ni0, nr1, ni1};
            __builtin_nontemporal_store(o, (f32x4*)op);   // write-once stream: TH=NT
            op += D_SZ * 2;
        }
    }
}

extern "C" void kernel_launch(void* const* d_in, const int* in_sizes, int n_in,
                              void* d_out, int out_size, void* d_ws, size_t ws_size,
                              hipStream_t stream) {
    const float* x  = (const float*)d_in[0];
    const float* Ar = (const float*)d_in[1];
    const float* Ai = (const float*)d_in[2];
    float* out = (float*)d_out;

    dim3 grid(B_SZ * (D_SZ / DBLK));   // 16 * 64 = 1024 blocks
    dim3 block(THREADS);               // 512 threads = 16 waves
    StatefulRecurrent_scan_kernel<<<grid, block, 0, stream>>>(x, Ar, Ai, out);
}